// compute_theta2_sparse_47321949667732
// MI455X (gfx1250) — compile-verified
//
#include <hip/hip_runtime.h>

#define N_PIXELS 262144
#define N_COMP   1000
#define NNZ      400000
#define N_BG     2
#define BATCH    128

typedef float v2f __attribute__((ext_vector_type(2)));
typedef float v8f __attribute__((ext_vector_type(8)));

// ---------------------------------------------------------------- zero output
__global__ void zero_out_kernel(float* __restrict__ Y, int n) {
    int i = blockIdx.x * blockDim.x + threadIdx.x;
    if (i < n) Y[i] = 0.0f;
}

// ------------------------------------------------- X[128,P] -> Xt[P,128] (f32)
// LDS tile transpose, 32x32 tiles, 32x8 threads, padded stride to kill bank
// conflicts. X is read exactly once -> non-temporal loads so the streamed
// 134 MB doesn't evict Xt (which the next two kernels re-read from L2).
__global__ void transpose_kernel(const float* __restrict__ X,
                                 float* __restrict__ Xt) {
    __shared__ float tile[32][33];
    const int bx = blockIdx.x;   // pixel tile   (P/32)
    const int by = blockIdx.y;   // batch tile   (128/32)
    const int tx = threadIdx.x;  // 0..31
    const int ty = threadIdx.y;  // 0..7
    const float* src = X + (size_t)(by * 32) * N_PIXELS + (size_t)bx * 32;
#pragma unroll
    for (int i = 0; i < 32; i += 8)
        tile[ty + i][tx] = __builtin_nontemporal_load(
            &src[(size_t)(ty + i) * N_PIXELS + tx]);
    __syncthreads();
    float* dst = Xt + (size_t)(bx * 32) * BATCH + (size_t)by * 32;
#pragma unroll
    for (int i = 0; i < 32; i += 8)
        dst[(size_t)(ty + i) * BATCH + tx] = tile[tx][ty + i];
}

// --------------------------------------------------- sparse AᵀXᵀ (COO gather)
// LDS-privatized [N_COMP x 32-batch] accumulator (128 KB of the 320 KB/WGP
// pool). grid = (nnz chunks, 4 batch quarters); 8 nnz streams x 32 batch lanes
// per 256-thread block. ds_add_f32 atomics in LDS, one coalesced
// global_atomic_add_f32 flush at the end. Look-ahead global_prefetch_b8 on the
// gathered Xt line hides the random-row L2 latency.
__global__ void spmm_kernel(const float* __restrict__ Xt,
                            const int*   __restrict__ rows,
                            const int*   __restrict__ cols,
                            const float* __restrict__ vals,
                            const float* __restrict__ nAtA,
                            float* __restrict__ Y,
                            int nnzPerBlock) {
    extern __shared__ float acc[];  // N_COMP * 32 floats = 128000 B
    const int tid = threadIdx.x;
    for (int i = tid; i < N_COMP * 32; i += 256) acc[i] = 0.0f;
    __syncthreads();

    const int quarter = blockIdx.y;      // 0..3 -> batch quarter
    const int lane    = tid & 31;        // batch lane within quarter
    const int ks      = tid >> 5;        // 0..7 -> nnz stream (one per wave)
    const float inv   = 1.0f / nAtA[0];
    const float* XtL  = Xt + quarter * 32 + lane;  // per-lane gather base

    int k0 = blockIdx.x * nnzPerBlock;
    int k1 = k0 + nnzPerBlock;
    if (k1 > NNZ) k1 = NNZ;

    for (int k = k0 + ks; k < k1; k += 8) {
        // software prefetch of a look-ahead gathered line (global_prefetch_b8)
        int kpf = k + 32;
        if (kpf < k1) {
            int rpf = rows[kpf];
            __builtin_prefetch(&XtL[(size_t)rpf * BATCH], 0, 1);
        }
        int   row = rows[k];               // uniform across the wave
        int   col = cols[k];
        float val = vals[k] * inv;
        float x   = XtL[(size_t)row * BATCH];        // coalesced 128B/wave
        atomicAdd(&acc[col * 32 + lane], val * x);   // ds_add_f32
    }
    __syncthreads();

    for (int i = tid; i < N_COMP * 32; i += 256) {
        int c = i >> 5, l = i & 31;
        atomicAdd(&Y[c * BATCH + quarter * 32 + l], acc[i]);
    }
}

// ------------------------------------------------ background bᵀXᵀ via WMMA f32
// One wave owns a 16(g,padded) x 16(batch) C tile; K=4 pixels per
// v_wmma_f32_16x16x4_f32 step. 8 waves/block cover all 128 batch columns,
// blocks partition the pixel axis, 2 live rows reduced with global atomics.
// Per-lane base pointers are hoisted and strided by compile-time constants so
// the hot loop is immediate-offset loads + one 64-bit add per stream, not
// per-iteration 64-bit address rebuilds. Loads are clamped + selected (no
// divergent branches) so EXEC stays all-ones for the WMMA.
__global__ void bg_wmma_kernel(const float* __restrict__ Xt,
                               const float* __restrict__ bgm,   // [P, 2]
                               const float* __restrict__ nAtA,
                               float* __restrict__ Y,
                               int pixPerBlock) {
    const int wave = threadIdx.x >> 5;
    const int lane = threadIdx.x & 31;
    const int b0   = wave * 16;          // batch column group
    const int m    = lane & 15;          // A row (g) / B col (batch)
    const int kb   = (lane >> 4) * 2;    // K sub-offset per half-wave
    const int p0   = blockIdx.x * pixPerBlock;
    const int iters = pixPerBlock >> 2;  // K=4 pixels per WMMA

    // Per-lane streaming pointers (constant byte strides inside the loop).
    const float* pA = bgm + (size_t)(p0 + kb) * N_BG + (m & 1);
    const float* pB = Xt  + (size_t)(p0 + kb) * BATCH + b0 + m;

    v8f c = {};
#pragma unroll 4
    for (int it = 0; it < iters; ++it) {
        // A fragment: 16x4 tile of b^T; rows m>=2 are zero via v_cndmask
        // (address clamped with m&1 so the load itself is unconditional).
        float g0 = pA[0];
        float g1 = pA[N_BG];
        v2f a, bb;
        a.x = (m < N_BG) ? g0 : 0.0f;
        a.y = (m < N_BG) ? g1 : 0.0f;
        // B fragment: 4x16 tile of Xt (pixels x batch)
        bb.x = pB[0];
        bb.y = pB[BATCH];
        c = __builtin_amdgcn_wmma_f32_16x16x4_f32(
                false, a, false, bb, (short)0, c, false, false);
        pA += 4 * N_BG;
        pB += 4 * BATCH;
    }

    const float inv = 1.0f / nAtA[0];
    // C layout: lanes 0-15 hold N=lane; VGPR v holds M=v (rows 0..7).
    // Only M=0,1 are real background rows.
    if (lane < 16) {
        atomicAdd(&Y[(N_COMP + 0) * BATCH + b0 + lane], c[0] * inv);
        atomicAdd(&Y[(N_COMP + 1) * BATCH + b0 + lane], c[1] * inv);
    }
}

// ------------------------------------------------------------------- launcher
extern "C" void kernel_launch(void* const* d_in, const int* in_sizes, int n_in,
                              void* d_out, int out_size, void* d_ws, size_t ws_size,
                              hipStream_t stream) {
    const float* X     = (const float*)d_in[0];   // [128, 262144]
    const float* Avals = (const float*)d_in[1];   // [400000]
    const float* bgm   = (const float*)d_in[2];   // [262144, 2]
    const float* nAtA  = (const float*)d_in[3];   // [1]
    const int*   Arows = (const int*)  d_in[4];   // [400000]
    const int*   Acols = (const int*)  d_in[5];   // [400000]
    float* Y  = (float*)d_out;                    // [1002, 128]
    float* Xt = (float*)d_ws;                     // needs P*128*4 = 128 MiB

    // 1) zero output (everything below accumulates via atomics)
    zero_out_kernel<<<(out_size + 255) / 256, 256, 0, stream>>>(Y, out_size);

    // 2) transpose X -> pixel-major Xt
    dim3 tb(32, 8), tg(N_PIXELS / 32, BATCH / 32);
    transpose_kernel<<<tg, tb, 0, stream>>>(X, Xt);

    // 3) sparse term into Y[0:1000]
    const int NCHUNK = 64;
    const int nnzPerBlock = (NNZ + NCHUNK - 1) / NCHUNK;
    dim3 sg(NCHUNK, 4);
    spmm_kernel<<<sg, 256, N_COMP * 32 * sizeof(float), stream>>>(
        Xt, Arows, Acols, Avals, nAtA, Y, nnzPerBlock);

    // 4) background term into Y[1000:1002] via WMMA
    const int PCHUNKS = 256;
    bg_wmma_kernel<<<PCHUNKS, 256, 0, stream>>>(
        Xt, bgm, nAtA, Y, N_PIXELS / PCHUNKS);
}